// NN_206158430333
// MI455X (gfx1250) — compile-verified
//
#include <hip/hip_runtime.h>
#include <math.h>

typedef float v2f __attribute__((ext_vector_type(2)));
typedef float v8f __attribute__((ext_vector_type(8)));

#define BATCH 8192
#define INHIB 0.01f
#define EPSV  1e-8f

// ---------- CDNA5 WMMA: D = A(16x4 f32) * B(4x16 f32) + C (16x16 f32) ----------
__device__ inline v8f wmma_f32_k4(v2f a, v2f b, v8f c) {
  // 8 args: (neg_a, A, neg_b, B, c_mod, C, reuse_a, reuse_b)
  return __builtin_amdgcn_wmma_f32_16x16x4_f32(false, a, false, b, (short)0, c,
                                               false, false);
}

// ---------- order-preserving float<->uint for atomic min/max ----------
__device__ inline unsigned ord_enc(float f) {
  unsigned u = __float_as_uint(f);
  return (u >> 31) ? ~u : (u | 0x80000000u);
}
__device__ inline float ord_dec(unsigned e) {
  return __uint_as_float((e >> 31) ? (e & 0x7fffffffu) : ~e);
}

// ---------- per-layer stat init (column sums + min/max sentinels) ----------
__global__ __launch_bounds__(256)
void k_init_stats(float* __restrict__ colsum, unsigned* __restrict__ stats) {
  for (int i = threadIdx.x; i < 2048; i += 256) colsum[i] = 0.f;
  if (threadIdx.x == 0) {
    stats[0] = 0xFFFFFFFFu; stats[1] = 0u;   // delta min / max
    stats[2] = 0xFFFFFFFFu; stats[3] = 0u;   // scale min / max
  }
}

// ---------- forward GEMM: Y[B x O] = X[B x K] @ W[O x K]^T + bias (opt relu) ----
// 128 threads = 4 waves. Block tile 64x64, K-stage 16. Wave w owns cols [16w,16w+16).
// Each wave holds 4 accumulators (all 64 rows x its 16 cols): A tile reused 4x.
__global__ __launch_bounds__(128)
void k_gemm_fwd(const float* __restrict__ X, const float* __restrict__ W,
                const float* __restrict__ bias, float* __restrict__ Y,
                int K, int O, int relu)
{
  __shared__ float lA[64 * 17];   // [m][k], padded stride kills bank conflicts
  __shared__ float lB[16 * 64];   // [k][n]
  const int tid  = threadIdx.x;
  const int wave = tid >> 5;
  const int lane = tid & 31;
  const int hi   = lane >> 4;
  const int ln   = lane & 15;
  const int rowBase = blockIdx.x * 64;
  const int colBase = blockIdx.y * 64;

  v8f acc[4] = {};

  const int ar = tid >> 1;          // 0..63 row of A tile
  const int ac = (tid & 1) * 8;     // 8-float chunk
  const int bn = tid >> 1;          // 0..63 col of B tile
  const int bk = (tid & 1) * 8;
  const int bcol = colBase + bn;
  const bool bok = bcol < O;

  for (int kk = 0; kk < K; kk += 16) {
    const float* xp = X + (size_t)(rowBase + ar) * K + kk + ac;
    #pragma unroll
    for (int j = 0; j < 8; ++j) lA[ar * 17 + ac + j] = xp[j];
    __builtin_prefetch(xp + 16, 0, 0);           // global_prefetch next K stage
    const float* wp = W + (size_t)bcol * K + kk + bk;
    #pragma unroll
    for (int j = 0; j < 8; ++j) lB[(bk + j) * 64 + bn] = bok ? wp[j] : 0.f;
    __syncthreads();
    #pragma unroll
    for (int k2 = 0; k2 < 16; k2 += 4) {
      v2f bf;
      bf.x = lB[(k2 + 2 * hi    ) * 64 + wave * 16 + ln];
      bf.y = lB[(k2 + 2 * hi + 1) * 64 + wave * 16 + ln];
      #pragma unroll
      for (int t = 0; t < 4; ++t) {
        v2f af;
        af.x = lA[(t * 16 + ln) * 17 + k2 + 2 * hi];
        af.y = lA[(t * 16 + ln) * 17 + k2 + 2 * hi + 1];
        acc[t] = wmma_f32_k4(af, bf, acc[t]);
      }
    }
    __syncthreads();
  }

  const int col = colBase + wave * 16 + ln;
  if (col < O) {
    const float bv = bias[col];
    #pragma unroll
    for (int t = 0; t < 4; ++t) {
      #pragma unroll
      for (int j = 0; j < 8; ++j) {
        int row = rowBase + t * 16 + hi * 8 + j;   // ISA C/D layout: M=j+8*hi
        float v = acc[t][j] + bv;
        if (relu) v = fmaxf(v, 0.f);
        Y[(size_t)row * O + col] = v;
      }
    }
  }
}

// ---------- inhibition + relu, one row per wave, shfl row-sum ----------
__global__ __launch_bounds__(256)
void k_inhibit(const float* __restrict__ Y, float* __restrict__ Yadj, int O)
{
  const int wave = threadIdx.x >> 5;
  const int lane = threadIdx.x & 31;
  const int row  = blockIdx.x * 8 + wave;
  const float* yr = Y + (size_t)row * O;
  float s = 0.f;
  for (int c = lane; c < O; c += 32) s += yr[c];
  #pragma unroll
  for (int m = 16; m > 0; m >>= 1) s += __shfl_xor(s, m, 32);
  float* ar = Yadj + (size_t)row * O;
  for (int c = lane; c < O; c += 32) {
    float yv = yr[c];
    ar[c] = fmaxf(yv - INHIB * (s - yv), 0.f);
  }
}

// ---------- column sum / sum-of-squares over the batch ----------
__global__ __launch_bounds__(256)
void k_colstats(const float* __restrict__ Yadj, float* __restrict__ msum,
                float* __restrict__ sqsum, int O)
{
  const int c = blockIdx.x * 256 + threadIdx.x;
  if (c >= O) return;
  float s = 0.f, s2 = 0.f;
  for (int r = blockIdx.y; r < BATCH; r += gridDim.y) {
    float v = Yadj[(size_t)r * O + c];
    s += v; s2 += v * v;
  }
  atomicAdd(&msum[c], s);
  atomicAdd(&sqsum[c], s2);
}

// ---------- Hebbian delta GEMM: D[O x N] = Yadj^T @ X - W*theta; + min/max ----
__global__ __launch_bounds__(128)
void k_hebb_delta(const float* __restrict__ Yadj, const float* __restrict__ X,
                  const float* __restrict__ W, const float* __restrict__ sqsum,
                  float* __restrict__ delta, unsigned* __restrict__ stats,
                  int O, int N)
{
  __shared__ float lA[64 * 17];   // [m][k] : transposed Yadj tile
  __shared__ float lB[16 * 64];   // [k][n] : X tile
  const int tid  = threadIdx.x;
  const int wave = tid >> 5;
  const int lane = tid & 31;
  const int hi   = lane >> 4;
  const int ln   = lane & 15;
  const int mBase = blockIdx.x * 64;
  const int nBase = blockIdx.y * 64;

  v8f acc[4] = {};

  const int akc = tid >> 3;        // k index 0..15
  const int amr = (tid & 7) * 8;   // m chunk start
  const int bkr = tid >> 3;        // k index 0..15
  const int bnc = (tid & 7) * 8;   // n chunk start

  for (int kk = 0; kk < BATCH; kk += 16) {
    const float* yp = Yadj + (size_t)(kk + akc) * O + mBase + amr;
    #pragma unroll
    for (int j = 0; j < 8; ++j) {
      int m = mBase + amr + j;
      lA[(amr + j) * 17 + akc] = (m < O) ? yp[j] : 0.f;
    }
    const float* xp = X + (size_t)(kk + bkr) * N + nBase + bnc;
    #pragma unroll
    for (int j = 0; j < 8; ++j) lB[bkr * 64 + bnc + j] = xp[j];
    __syncthreads();
    #pragma unroll
    for (int k2 = 0; k2 < 16; k2 += 4) {
      v2f bf;
      bf.x = lB[(k2 + 2 * hi    ) * 64 + wave * 16 + ln];
      bf.y = lB[(k2 + 2 * hi + 1) * 64 + wave * 16 + ln];
      #pragma unroll
      for (int t = 0; t < 4; ++t) {
        v2f af;
        af.x = lA[(t * 16 + ln) * 17 + k2 + 2 * hi];
        af.y = lA[(t * 16 + ln) * 17 + k2 + 2 * hi + 1];
        acc[t] = wmma_f32_k4(af, bf, acc[t]);
      }
    }
    __syncthreads();
  }

  const int n = nBase + wave * 16 + ln;
  float lmin = 3.4e38f, lmax = -3.4e38f;
  #pragma unroll
  for (int t = 0; t < 4; ++t) {
    #pragma unroll
    for (int j = 0; j < 8; ++j) {
      int m = mBase + t * 16 + hi * 8 + j;
      if (m < O) {
        float th = sqsum[m] * (1.0f / BATCH);        // theta = mean(y^2)
        float d = acc[t][j] - W[(size_t)m * N + n] * th;   // LR = 1
        delta[(size_t)m * N + n] = d;
        lmin = fminf(lmin, d); lmax = fmaxf(lmax, d);
      }
    }
  }
  #pragma unroll
  for (int m = 16; m > 0; m >>= 1) {
    lmin = fminf(lmin, __shfl_xor(lmin, m, 32));
    lmax = fmaxf(lmax, __shfl_xor(lmax, m, 32));
  }
  if (lane == 0) {
    atomicMin(&stats[0], ord_enc(lmin));
    atomicMax(&stats[1], ord_enc(lmax));
  }
}

// ---------- in-place (v - min)/(max - min + eps) ----------
__global__ __launch_bounds__(256)
void k_norm(float* __restrict__ buf, const unsigned* __restrict__ stats, int count)
{
  int i = blockIdx.x * 256 + threadIdx.x;
  if (i >= count) return;
  float mn = ord_dec(stats[0]);
  float mx = ord_dec(stats[1]);
  buf[i] = (buf[i] - mn) / (mx - mn + EPSV);
}

// ---------- stable rank of -y_mean: masks, losers, winner flags ----------
__global__ __launch_bounds__(256)
void k_rank(const float* __restrict__ msum, int O, int nwin,
            float* __restrict__ mask_out, float* __restrict__ loser_out,
            int* __restrict__ wflag)
{
  __shared__ float sv[1024];
  for (int i = threadIdx.x; i < O; i += 256) sv[i] = msum[i];  // positive scale of mean
  __syncthreads();
  for (int j = threadIdx.x; j < O; j += 256) {
    float v = sv[j];
    int rank = 0;
    for (int k = 0; k < O; ++k) {
      float vk = sv[k];
      rank += (vk > v) || (vk == v && k < j);   // stable argsort of -mean
    }
    int win = rank < nwin;
    mask_out[j] = win ? 1.f : 0.f;
    wflag[j] = win;
    if (!win && rank <= O - 2) loser_out[rank - nwin] = (float)j;  // order[nwin:-1]
  }
}

// ---------- scale = winner-masked delta; gather min/max ----------
__global__ __launch_bounds__(256)
void k_scale_build(const float* __restrict__ delta, const int* __restrict__ wflag,
                   float* __restrict__ sout, unsigned* __restrict__ stats,
                   int count, int N)
{
  int i = blockIdx.x * 256 + threadIdx.x;
  bool ok = i < count;
  float v = 0.f;
  if (ok) {
    int m = i / N;
    v = wflag[m] ? delta[i] : 0.f;
    sout[i] = v;
  }
  float lmin = ok ? v : 3.4e38f;
  float lmax = ok ? v : -3.4e38f;
  #pragma unroll
  for (int m = 16; m > 0; m >>= 1) {
    lmin = fminf(lmin, __shfl_xor(lmin, m, 32));
    lmax = fmaxf(lmax, __shfl_xor(lmax, m, 32));
  }
  if ((threadIdx.x & 31) == 0) {
    atomicMin(&stats[0], ord_enc(lmin));
    atomicMax(&stats[1], ord_enc(lmax));
  }
}

// ---------- row softmax over final logits ----------
__global__ __launch_bounds__(256)
void k_softmax(const float* __restrict__ Yl, float* __restrict__ P, int O)
{
  __shared__ float red[256];
  const int row = blockIdx.x;
  const float* yr = Yl + (size_t)row * O;
  float mx = -3.4e38f;
  for (int c = threadIdx.x; c < O; c += 256) mx = fmaxf(mx, yr[c]);
  red[threadIdx.x] = mx;
  __syncthreads();
  for (int s = 128; s > 0; s >>= 1) {
    if (threadIdx.x < s) red[threadIdx.x] = fmaxf(red[threadIdx.x], red[threadIdx.x + s]);
    __syncthreads();
  }
  mx = red[0];
  __syncthreads();
  float sum = 0.f;
  for (int c = threadIdx.x; c < O; c += 256) sum += expf(yr[c] - mx);
  red[threadIdx.x] = sum;
  __syncthreads();
  for (int s = 128; s > 0; s >>= 1) {
    if (threadIdx.x < s) red[threadIdx.x] += red[threadIdx.x + s];
    __syncthreads();
  }
  float inv = 1.0f / red[0];
  for (int c = threadIdx.x; c < O; c += 256) P[(size_t)row * O + c] = expf(yr[c] - mx) * inv;
}

extern "C" void kernel_launch(void* const* d_in, const int* in_sizes, int n_in,
                              void* d_out, int out_size, void* d_ws, size_t ws_size,
                              hipStream_t stream)
{
  (void)in_sizes; (void)n_in; (void)out_size; (void)ws_size;
  const float* x0 = (const float*)d_in[0];
  const float* W[4] = {(const float*)d_in[1], (const float*)d_in[3],
                       (const float*)d_in[5], (const float*)d_in[7]};
  const float* b[4] = {(const float*)d_in[2], (const float*)d_in[4],
                       (const float*)d_in[6], (const float*)d_in[8]};
  const int fan[4] = {8192, 256, 128, 64};
  const int osz[4] = {256, 128, 64, 1000};

  // workspace layout (floats): ~89 MB total
  float* wsf = (float*)d_ws;
  size_t off = 0;
  float* x1    = wsf + off; off += (size_t)8192 * 256;
  float* x2    = wsf + off; off += (size_t)8192 * 128;
  float* x3    = wsf + off; off += (size_t)8192 * 64;
  float* y3    = wsf + off; off += (size_t)8192 * 1000;
  float* yadj  = wsf + off; off += (size_t)8192 * 1000;
  float* delta = wsf + off; off += (size_t)256 * 8192;
  float* msum  = wsf + off; off += 1024;        // contiguous with sqsum: zeroed together
  float* sqsum = wsf + off; off += 1024;
  unsigned* stats = (unsigned*)(wsf + off); off += 4;
  int* wflag = (int*)(wsf + off); off += 1024;

  const float* xin[4] = {x0, x1, x2, x3};
  float* xout[4]      = {x1, x2, x3, y3};

  // d_out layout: probs | scales 0..3 | losers 0..3 | masks 0..3
  float* out = (float*)d_out;
  size_t o = 0;
  float* probs = out + o; o += (size_t)8192 * 1000;
  float* sc[4];
  sc[0] = out + o; o += (size_t)256 * 8192;
  sc[1] = out + o; o += (size_t)128 * 256;
  sc[2] = out + o; o += (size_t)64 * 128;
  sc[3] = out + o; o += (size_t)1000 * 64;
  float* ls[4];
  ls[0] = out + o; o += 127;
  ls[1] = out + o; o += 63;
  ls[2] = out + o; o += 31;
  ls[3] = out + o; o += 499;
  float* mk[4];
  mk[0] = out + o; o += 256;
  mk[1] = out + o; o += 128;
  mk[2] = out + o; o += 64;
  mk[3] = out + o; o += 1000;

  for (int i = 0; i < 4; ++i) {
    const int O = osz[i], N = fan[i];
    const int nwin = O / 2;
    const int cnt = O * N;
    k_init_stats<<<1, 256, 0, stream>>>(msum, stats);
    dim3 g1(8192 / 64, (O + 63) / 64);
    k_gemm_fwd<<<g1, 128, 0, stream>>>(xin[i], W[i], b[i], xout[i], N, O, (i < 3) ? 1 : 0);
    k_inhibit<<<8192 / 8, 256, 0, stream>>>(xout[i], yadj, O);
    dim3 g2((O + 255) / 256, 32);
    k_colstats<<<g2, 256, 0, stream>>>(yadj, msum, sqsum, O);
    dim3 g3((O + 63) / 64, N / 64);
    k_hebb_delta<<<g3, 128, 0, stream>>>(yadj, xin[i], W[i], sqsum, delta, stats, O, N);
    k_norm<<<(cnt + 255) / 256, 256, 0, stream>>>(delta, stats, cnt);
    k_rank<<<1, 256, 0, stream>>>(msum, O, nwin, mk[i], ls[i], wflag);
    k_scale_build<<<(cnt + 255) / 256, 256, 0, stream>>>(delta, wflag, sc[i], stats + 2, cnt, N);
    k_norm<<<(cnt + 255) / 256, 256, 0, stream>>>(sc[i], stats + 2, cnt);
  }
  k_softmax<<<8192, 256, 0, stream>>>(y3, probs, 1000);
}